// twoD_RNN_5892695130588
// MI455X (gfx1250) — compile-verified
//
#include <hip/hip_runtime.h>
#include <math.h>

// 2D neural-field RNN, 64x64 torus, T=4000 serial steps.
// Circular Gaussian conv == circulant GEMM:  g = M * r * M^T  (M 64x64, constant).
// One persistent workgroup (32 wave32 waves) keeps everything in LDS; the 4
// per-step 64^3 GEMMs run on V_WMMA_F32_16X16X4_F32 (f32 preserves the iterated
// dynamics over 4000 steps; problem is latency-bound, not throughput-bound).

#define NN      64
#define TSTEPS  4000
#define KS      63
#define LDSP    65          // padded LDS row stride (bank-conflict-free rows)

typedef float v2f __attribute__((ext_vector_type(2)));
typedef float v8f __attribute__((ext_vector_type(8)));

__device__ __forceinline__ float phi_act(float x) {
    return x > 0.0f ? tanhf(x) : 0.0f;
}

__launch_bounds__(1024, 1)
__global__ void twoD_RNN_persistent(const float* __restrict__ ipt,       // [N,N,T]
                                    const float* __restrict__ log_sig_e, // [1]
                                    const float* __restrict__ log_sig_i, // [1]
                                    const float* __restrict__ w_out_g,   // [N*N,1]
                                    const float* __restrict__ he0,       // [N,N]
                                    const float* __restrict__ hi0,
                                    const float* __restrict__ re0,
                                    const float* __restrict__ ri0,
                                    float* __restrict__ out)             // re|ri|readout
{
    // LDS: 0 = excitatory path, 1 = inhibitory path
    __shared__ float Mb[2][NN][LDSP];   // circulant matrices Me, Mi (constant)
    __shared__ float Rb[2][NN][LDSP];   // rates re, ri
    __shared__ float Ub[2][NN][LDSP];   // stage1 result U = M*r, then G = U*M^T in place
    __shared__ float WOUT[NN * NN];
    __shared__ float tapE[NN];          // 1D gaussian taps, index 63 == 0 (missing tap)
    __shared__ float tapI[NN];
    __shared__ float partial[32];

    const int tid  = threadIdx.x;
    const int lane = tid & 31;
    const int wv   = tid >> 5;          // 0..31
    const int half = lane >> 4;         // 0 | 1
    const int l16  = lane & 15;

    // ---------------- prologue: taps, state, weights ----------------
    if (tid < NN) {
        float ve = 0.0f, vi = 0.0f;
        if (tid < KS) {
            const float c  = (KS - 1) * 0.5f;              // 31
            const float d  = (float)tid - c;
            const float se = expf(log_sig_e[0]) * (float)KS;
            const float si = expf(log_sig_i[0]) * (float)KS;
            ve = expf(-(d * d) / (2.0f * se * se));
            vi = expf(-(d * d) / (2.0f * si * si));
        }
        tapE[tid] = ve;
        tapI[tid] = vi;
    }
    float hE[4], hI[4];
    #pragma unroll
    for (int c = 0; c < 4; ++c) {
        const int idx = tid + c * 1024;
        const int i = idx >> 6, j = idx & 63;
        hE[c] = he0[idx];
        hI[c] = hi0[idx];
        Rb[0][i][j] = re0[idx];
        Rb[1][i][j] = ri0[idx];
        WOUT[idx] = w_out_g[idx];
    }
    __syncthreads();

    // normalize taps, build circulant matrices: M[i][m] = w[(m-i+31) & 63], w[63]=0
    float sE = 0.0f, sI = 0.0f;
    for (int a = 0; a < KS; ++a) { sE += tapE[a]; sI += tapI[a]; }
    const float invE = 1.0f / sE, invI = 1.0f / sI;
    #pragma unroll
    for (int c = 0; c < 4; ++c) {
        const int idx = tid + c * 1024;
        const int i = idx >> 6, m = idx & 63;
        const int a = (m - i + 31) & 63;
        Mb[0][i][m] = tapE[a] * invE;
        Mb[1][i][m] = tapI[a] * invI;
    }
    __syncthreads();

    // scalar constants (match reference)
    const float SQPI = 1.7724538509055160f;                 // sqrt(pi)
    const float WEE  =  1.00f * 64.0f * 0.1f * SQPI;
    const float WEI  = -2.00f * 64.0f * 0.2f * SQPI;
    const float WIE  =  0.99f * 64.0f * 0.1f * SQPI;
    const float WII  = -1.80f * 64.0f * 0.2f * SQPI;
    const float MU_E = 1.0f, MU_I = 0.8f;
    const float AE   = 0.2f;                                 // DT/TAU_E
    const float AI   = (float)(0.001 / 0.015);               // DT/TAU_I

    const long long offRI = (long long)NN * NN * TSTEPS;     // ri field offset
    const long long offRO = 2ll * offRI;                     // readout offset

    // per-wave WMMA tile: waves 0-15 -> e path, 16-31 -> i path; 16 tiles each
    const int p  = wv >> 4;              // population
    const int tt = wv & 15;
    const int i0 = (tt >> 2) * 16;
    const int j0 = (tt & 3) * 16;

    // ---------------- serial scan over time ----------------
    for (int t = 0; t < TSTEPS; ++t) {
        // issue this step's input loads early (overlap with GEMMs)
        float x[4];
        #pragma unroll
        for (int c = 0; c < 4; ++c)
            x[c] = ipt[(long long)(tid + c * 1024) * TSTEPS + t];

        // ---- stage 1: U = M * r  (16x16 tile, K=64 via 16 x K=4 WMMA) ----
        {
            v8f acc = {};
            #pragma unroll
            for (int k0 = 0; k0 < NN; k0 += 4) {
                const int kk = k0 + 2 * half;
                v2f a, b;
                a.x = Mb[p][i0 + l16][kk];
                a.y = Mb[p][i0 + l16][kk + 1];
                b.x = Rb[p][kk][j0 + l16];
                b.y = Rb[p][kk + 1][j0 + l16];
                acc = __builtin_amdgcn_wmma_f32_16x16x4_f32(
                          false, a, false, b, (short)0, acc, false, false);
            }
            #pragma unroll
            for (int v = 0; v < 8; ++v)
                Ub[p][i0 + half * 8 + v][j0 + l16] = acc[v];
        }
        __syncthreads();                       // U complete

        // ---- stage 2: G = U * M^T ; B[k][j] = M[j][k] -> row-major reads ----
        v8f acc2 = {};
        #pragma unroll
        for (int k0 = 0; k0 < NN; k0 += 4) {
            const int kk = k0 + 2 * half;
            v2f a, b;
            a.x = Ub[p][i0 + l16][kk];
            a.y = Ub[p][i0 + l16][kk + 1];
            b.x = Mb[p][j0 + l16][kk];
            b.y = Mb[p][j0 + l16][kk + 1];
            acc2 = __builtin_amdgcn_wmma_f32_16x16x4_f32(
                       false, a, false, b, (short)0, acc2, false, false);
        }
        __syncthreads();                       // everyone finished reading U
        #pragma unroll
        for (int v = 0; v < 8; ++v)
            Ub[p][i0 + half * 8 + v][j0 + l16] = acc2[v];   // G in place over U
        __syncthreads();                       // G visible

        // ---- pointwise update, outputs, readout partial ----
        float acc_ro = 0.0f;
        #pragma unroll
        for (int c = 0; c < 4; ++c) {
            const int idx = tid + c * 1024;
            const int i = idx >> 6, j = idx & 63;
            const float ge = Ub[0][i][j];
            const float gi = Ub[1][i][j];
            const float de = WEE * ge + WEI * gi + MU_E;
            const float di = WIE * ge + WII * gi + MU_I;
            hE[c] += AE * (-hE[c] + de + x[c]);     // W_in == 1
            hI[c] += AI * (-hI[c] + di);
            const float re = phi_act(hE[c]);
            const float ri = phi_act(hI[c]);
            Rb[0][i][j] = re;
            Rb[1][i][j] = ri;
            out[(long long)idx * TSTEPS + t]         = re;   // re_xy [N,N,T]
            out[offRI + (long long)idx * TSTEPS + t] = ri;   // ri_xy [N,N,T]
            acc_ro += re * WOUT[idx];
        }
        // wave32 reduction, then per-wave partial -> thread 0
        #pragma unroll
        for (int s = 16; s > 0; s >>= 1)
            acc_ro += __shfl_down(acc_ro, s, 32);
        if (lane == 0) partial[wv] = acc_ro;
        __syncthreads();                       // rates + partials visible
        if (tid == 0) {
            float s = 0.0f;
            #pragma unroll
            for (int w = 0; w < 32; ++w) s += partial[w];
            out[offRO + t] = s;                // readout [1,T]
        }
    }
}

extern "C" void kernel_launch(void* const* d_in, const int* in_sizes, int n_in,
                              void* d_out, int out_size, void* d_ws, size_t ws_size,
                              hipStream_t stream) {
    (void)in_sizes; (void)n_in; (void)out_size; (void)d_ws; (void)ws_size;
    const float* ipt       = (const float*)d_in[0];
    const float* log_sig_e = (const float*)d_in[1];
    const float* log_sig_i = (const float*)d_in[2];
    const float* w_out     = (const float*)d_in[3];
    const float* he0       = (const float*)d_in[4];
    const float* hi0       = (const float*)d_in[5];
    const float* re0       = (const float*)d_in[6];
    const float* ri0       = (const float*)d_in[7];
    float* out             = (float*)d_out;

    twoD_RNN_persistent<<<1, 1024, 0, stream>>>(
        ipt, log_sig_e, log_sig_i, w_out, he0, hi0, re0, ri0, out);
}